// MyBertMainWordPiece_43267500540792
// MI455X (gfx1250) — compile-verified
//
#include <hip/hip_runtime.h>

typedef __attribute__((ext_vector_type(2))) float v2f;
typedef __attribute__((ext_vector_type(8))) float v8f;

#define HDIM   768
#define SEQ    128
#define NWORDS 51
#define SLOT   122
#define INTENT 22
#define BATCH  512

#define KC       192          // K-chunk staged in LDS
#define LDSTR    196          // KC + 4 pad: 196 % 64 == 4 -> conflict-free fragment reads

// ---------------------------------------------------------------------------
// Slot head: gather/mean rows of hidden -> LDS, then fp32 WMMA GEMM vs W_slot.
// grid.x = (BATCH*NWORDS)/32 = 816 blocks, 256 threads (8 waves).
// Each block: 32 output rows; wave w owns N-tile w (cols 16w..16w+15).
// ---------------------------------------------------------------------------
__global__ __launch_bounds__(256) void slot_head_kernel(
    const float* __restrict__ hidden,   // [B, SEQ, H]
    const float* __restrict__ W_slot,   // [H, SLOT]
    const float* __restrict__ b_slot,   // [SLOT]
    const int*   __restrict__ tsi,      // [B, NWORDS]
    const int*   __restrict__ sub,      // [B, NWORDS]
    float*       __restrict__ out)      // [B*NWORDS, SLOT]
{
  __shared__ float lds[32 * LDSTR];

  const int tid   = threadIdx.x;
  const int lane  = tid & 31;
  const int wave  = tid >> 5;
  const int m0    = blockIdx.x * 32;

  const int row16 = lane & 15;          // fragment row (A) / col (B)
  const int koff  = (lane >> 4) << 1;   // lanes 16..31 hold K=2,3
  const int n     = wave * 16 + row16;  // output column this lane owns
  const bool nvalid = (n < SLOT);
  const int  nload  = nvalid ? n : (SLOT - 1);   // clamp for safe loads

  // C init = bias broadcast (garbage in clamped columns is never stored)
  const float bias = b_slot[nload];
  v8f c0, c1;
#pragma unroll
  for (int r = 0; r < 8; ++r) { c0[r] = bias; c1[r] = bias; }

  for (int kc = 0; kc < HDIM; kc += KC) {
    // ---- cooperative gather + subword-mean into LDS (coalesced) ----
    for (int idx = tid; idx < 32 * KC; idx += 256) {
      const int r  = idx / KC;
      const int c  = idx - r * KC;
      const int gm = m0 + r;
      const int b  = gm / NWORDS;
      const int w  = gm - b * NWORDS;
      const int start = tsi[b * NWORDS + w];
      const int len   = sub[b * NWORDS + w];
      const float* base = hidden + ((size_t)b * SEQ + start) * HDIM + kc + c;
      float v = base[0];
      if (len != 1) v = 0.5f * (v + base[(size_t)len * HDIM]);  // mean of {start, start+len}
      lds[r * LDSTR + c] = v;
    }
    __syncthreads();

    // ---- fp32 WMMA k-loop: D = A x B + C, 16x16x4 ----
#pragma unroll 4
    for (int k = 0; k < KC; k += 4) {
      v2f a0, a1, bf;
      const int la = row16 * LDSTR + k + koff;
      a0.x = lds[la];
      a0.y = lds[la + 1];
      const int lb = (row16 + 16) * LDSTR + k + koff;
      a1.x = lds[lb];
      a1.y = lds[lb + 1];
      const float* wp = W_slot + (size_t)(kc + k + koff) * SLOT + nload;
      bf.x = wp[0];
      bf.y = wp[SLOT];
      c0 = __builtin_amdgcn_wmma_f32_16x16x4_f32(false, a0, false, bf,
                                                 (short)0, c0, false, false);
      c1 = __builtin_amdgcn_wmma_f32_16x16x4_f32(false, a1, false, bf,
                                                 (short)0, c1, false, false);
    }
    __syncthreads();
  }

  // ---- store: VGPR r -> M=r (lanes 0-15) / M=r+8 (lanes 16-31) ----
  if (nvalid) {
    const int rbase = (lane >> 4) << 3;
#pragma unroll
    for (int r = 0; r < 8; ++r) {
      const int gr = m0 + rbase + r;
      out[(size_t)gr * SLOT + n]        = c0[r];
      out[(size_t)(gr + 16) * SLOT + n] = c1[r];
    }
  }
}

// ---------------------------------------------------------------------------
// Intent head: pooled @ W_id + b_id.  grid.x = BATCH/16 = 32 blocks,
// 64 threads (2 waves), wave w owns N-tile w (cols 16w..16w+15; only 22 valid).
// ---------------------------------------------------------------------------
__global__ __launch_bounds__(64) void intent_head_kernel(
    const float* __restrict__ pooled,   // [B, H]
    const float* __restrict__ W_id,     // [H, INTENT]
    const float* __restrict__ b_id,     // [INTENT]
    float*       __restrict__ out)      // [B, INTENT]
{
  __shared__ float lds[16 * LDSTR];

  const int tid   = threadIdx.x;
  const int lane  = tid & 31;
  const int wave  = tid >> 5;
  const int m0    = blockIdx.x * 16;

  const int row16 = lane & 15;
  const int koff  = (lane >> 4) << 1;
  const int n     = wave * 16 + row16;
  const bool nvalid = (n < INTENT);
  const int  nload  = nvalid ? n : (INTENT - 1);

  const float bias = b_id[nload];
  v8f c0;
#pragma unroll
  for (int r = 0; r < 8; ++r) c0[r] = bias;

  for (int kc = 0; kc < HDIM; kc += KC) {
    for (int idx = tid; idx < 16 * KC; idx += 64) {
      const int r = idx / KC;
      const int c = idx - r * KC;
      lds[r * LDSTR + c] = pooled[(size_t)(m0 + r) * HDIM + kc + c];
    }
    __syncthreads();

#pragma unroll 4
    for (int k = 0; k < KC; k += 4) {
      v2f a0, bf;
      const int la = row16 * LDSTR + k + koff;
      a0.x = lds[la];
      a0.y = lds[la + 1];
      const float* wp = W_id + (size_t)(kc + k + koff) * INTENT + nload;
      bf.x = wp[0];
      bf.y = wp[INTENT];
      c0 = __builtin_amdgcn_wmma_f32_16x16x4_f32(false, a0, false, bf,
                                                 (short)0, c0, false, false);
    }
    __syncthreads();
  }

  if (nvalid) {
    const int rbase = (lane >> 4) << 3;
#pragma unroll
    for (int r = 0; r < 8; ++r)
      out[(size_t)(m0 + rbase + r) * INTENT + n] = c0[r];
  }
}

// ---------------------------------------------------------------------------
extern "C" void kernel_launch(void* const* d_in, const int* in_sizes, int n_in,
                              void* d_out, int out_size, void* d_ws, size_t ws_size,
                              hipStream_t stream) {
  const float* hidden = (const float*)d_in[0];
  const float* pooled = (const float*)d_in[1];
  const float* W_id   = (const float*)d_in[2];
  const float* b_id   = (const float*)d_in[3];
  const float* W_slot = (const float*)d_in[4];
  const float* b_slot = (const float*)d_in[5];
  const int*   tsi    = (const int*)d_in[6];
  const int*   sub    = (const int*)d_in[7];

  float* out_id = (float*)d_out;                  // [512, 22]
  float* out_sf = out_id + (size_t)BATCH * INTENT; // [512*51, 122]

  intent_head_kernel<<<BATCH / 16, 64, 0, stream>>>(pooled, W_id, b_id, out_id);
  slot_head_kernel<<<(BATCH * NWORDS) / 32, 256, 0, stream>>>(
      hidden, W_slot, b_slot, tsi, sub, out_sf);
}